// QCNNQLSTM_QSampler_65481071399483
// MI455X (gfx1250) — compile-verified
//
#include <hip/hip_runtime.h>
#include <math.h>

typedef float v2f __attribute__((ext_vector_type(2)));
typedef float v8f __attribute__((ext_vector_type(8)));

#define DEVINL __device__ __forceinline__

#define NBATCH 4096
#define PI_F 3.14159265358979323846f

DEVINL float shflx(float v, int mask) { return __shfl_xor(v, mask, 32); }

// ---------------------------------------------------------------------------
// 8-qubit statevector: amp index a = lane*8 + r  (a in [0,256))
// qubit q lives at bit p = 7-q of a:  q0..q4 -> lane bits 4..0, q5..q7 -> r bits 2..0
// ---------------------------------------------------------------------------

template<int Q>
DEVINL void gH(float (&re)[8], float (&im)[8], int lane) {
  const float K = 0.70710678118654752440f;
  if constexpr (Q >= 5) {
    constexpr int rb = 1 << (7 - Q);
    #pragma unroll
    for (int r0 = 0; r0 < 8; ++r0) {
      if ((r0 & rb) == 0) {
        const int r1 = r0 | rb;
        float a0r = re[r0], a0i = im[r0], a1r = re[r1], a1i = im[r1];
        re[r0] = (a0r + a1r) * K; im[r0] = (a0i + a1i) * K;
        re[r1] = (a0r - a1r) * K; im[r1] = (a0i - a1i) * K;
      }
    }
  } else {
    constexpr int lb = 1 << (4 - Q);
    const bool hi = (lane & lb) != 0;
    #pragma unroll
    for (int r = 0; r < 8; ++r) {
      float orr = shflx(re[r], lb), oii = shflx(im[r], lb);
      re[r] = (hi ? (orr - re[r]) : (re[r] + orr)) * K;
      im[r] = (hi ? (oii - im[r]) : (im[r] + oii)) * K;
    }
  }
}

// rz(t) = diag(e^{-it/2}, e^{+it/2})  (diagonal: no shuffles)
template<int Q>
DEVINL void gRZ(float (&re)[8], float (&im)[8], float t, int lane) {
  float s, c; sincosf(0.5f * t, &s, &c);
  constexpr int p = 7 - Q;
  #pragma unroll
  for (int r = 0; r < 8; ++r) {
    bool bit;
    if constexpr (p < 3) bit = ((r >> p) & 1) != 0;
    else                 bit = ((lane >> (p - 3)) & 1) != 0;
    float ms = bit ? s : -s;
    float nr = re[r] * c - im[r] * ms;
    float ni = re[r] * ms + im[r] * c;
    re[r] = nr; im[r] = ni;
  }
}

// phase(lam) = diag(1, e^{i lam})  (diagonal)
template<int Q>
DEVINL void gPH(float (&re)[8], float (&im)[8], float lam, int lane) {
  float s, c; sincosf(lam, &s, &c);
  constexpr int p = 7 - Q;
  #pragma unroll
  for (int r = 0; r < 8; ++r) {
    bool bit;
    if constexpr (p < 3) bit = ((r >> p) & 1) != 0;
    else                 bit = ((lane >> (p - 3)) & 1) != 0;
    float nr = re[r] * c - im[r] * s;
    float ni = re[r] * s + im[r] * c;
    re[r] = bit ? nr : re[r];
    im[r] = bit ? ni : im[r];
  }
}

// ry(t) = [[c,-s],[s,c]]  (real)
template<int Q>
DEVINL void gRY(float (&re)[8], float (&im)[8], float t, int lane) {
  float s, c; sincosf(0.5f * t, &s, &c);
  constexpr int p = 7 - Q;
  if constexpr (p < 3) {
    constexpr int rb = 1 << p;
    #pragma unroll
    for (int r0 = 0; r0 < 8; ++r0) {
      if ((r0 & rb) == 0) {
        const int r1 = r0 | rb;
        float a0r = re[r0], a0i = im[r0], a1r = re[r1], a1i = im[r1];
        re[r0] = c * a0r - s * a1r; im[r0] = c * a0i - s * a1i;
        re[r1] = s * a0r + c * a1r; im[r1] = s * a0i + c * a1i;
      }
    }
  } else {
    constexpr int lb = 1 << (p - 3);
    const bool hi = (lane & lb) != 0;
    #pragma unroll
    for (int r = 0; r < 8; ++r) {
      float orr = shflx(re[r], lb), oii = shflx(im[r], lb);
      re[r] = hi ? (s * orr + c * re[r]) : (c * re[r] - s * orr);
      im[r] = hi ? (s * oii + c * im[r]) : (c * im[r] - s * oii);
    }
  }
}

// CNOT control C, target T
template<int C, int T>
DEVINL void gCN(float (&re)[8], float (&im)[8], int lane) {
  constexpr int pc = 7 - C, pt = 7 - T;
  if constexpr (pc < 3 && pt < 3) {
    constexpr int cb = 1 << pc, tb = 1 << pt;
    #pragma unroll
    for (int r0 = 0; r0 < 8; ++r0) {
      if ((r0 & cb) && !(r0 & tb)) {
        const int r1 = r0 | tb;
        float tr = re[r0]; re[r0] = re[r1]; re[r1] = tr;
        float ti = im[r0]; im[r0] = im[r1]; im[r1] = ti;
      }
    }
  } else if constexpr (pc >= 3 && pt < 3) {
    constexpr int cb = 1 << (pc - 3), tb = 1 << pt;
    const bool ctl = (lane & cb) != 0;
    #pragma unroll
    for (int r0 = 0; r0 < 8; ++r0) {
      if (!(r0 & tb)) {
        const int r1 = r0 | tb;
        float ar = re[r0], br = re[r1], ai = im[r0], bi = im[r1];
        re[r0] = ctl ? br : ar; re[r1] = ctl ? ar : br;
        im[r0] = ctl ? bi : ai; im[r1] = ctl ? ai : bi;
      }
    }
  } else if constexpr (pc < 3 && pt >= 3) {
    constexpr int cb = 1 << pc, tb = 1 << (pt - 3);
    #pragma unroll
    for (int r = 0; r < 8; ++r) {
      if (r & cb) {  // both shuffle partners share r, so swap unconditionally
        re[r] = shflx(re[r], tb);
        im[r] = shflx(im[r], tb);
      }
    }
  } else {
    constexpr int cb = 1 << (pc - 3), tb = 1 << (pt - 3);
    const bool ctl = (lane & cb) != 0;
    #pragma unroll
    for (int r = 0; r < 8; ++r) {
      float orr = shflx(re[r], tb), oii = shflx(im[r], tb);
      re[r] = ctl ? orr : re[r];
      im[r] = ctl ? oii : im[r];
    }
  }
}

template<int A, int B>
DEVINL void convblk(float (&re)[8], float (&im)[8], const float* p, int lane) {
  gRZ<B>(re, im, -0.5f * PI_F, lane);
  gCN<B, A>(re, im, lane);
  gRZ<A>(re, im, p[0], lane);
  gRY<B>(re, im, p[1], lane);
  gCN<A, B>(re, im, lane);
  gRY<B>(re, im, p[2], lane);
  gCN<B, A>(re, im, lane);
  gRZ<A>(re, im, 0.5f * PI_F, lane);
}

template<int A, int B>
DEVINL void poolblk(float (&re)[8], float (&im)[8], const float* p, int lane) {
  gRZ<B>(re, im, -0.5f * PI_F, lane);
  gCN<B, A>(re, im, lane);
  gRZ<A>(re, im, p[0], lane);
  gRY<B>(re, im, p[1], lane);
  gCN<A, B>(re, im, lane);
  gRY<B>(re, im, p[2], lane);
}

// One wave per batch element: 4096 waves, each simulates the full circuit.
__global__ void k_qcnn(const float* __restrict__ inp,
                       const float* __restrict__ c1, const float* __restrict__ p1,
                       const float* __restrict__ c2, const float* __restrict__ p2,
                       const float* __restrict__ c3, const float* __restrict__ p3,
                       float* __restrict__ qout) {
  const int wid  = (blockIdx.x * blockDim.x + threadIdx.x) >> 5;
  const int lane = threadIdx.x & 31;
  if (wid >= NBATCH) return;

  float x[8];
  #pragma unroll
  for (int q = 0; q < 8; ++q) x[q] = inp[wid * 8 + q];

  float re[8], im[8];
  #pragma unroll
  for (int r = 0; r < 8; ++r) { re[r] = 0.0f; im[r] = 0.0f; }
  re[0] = (lane == 0) ? 1.0f : 0.0f;

  #pragma unroll
  for (int rep = 0; rep < 2; ++rep) {
    gH<0>(re, im, lane); gH<1>(re, im, lane); gH<2>(re, im, lane); gH<3>(re, im, lane);
    gH<4>(re, im, lane); gH<5>(re, im, lane); gH<6>(re, im, lane); gH<7>(re, im, lane);
    gPH<0>(re, im, 2.0f * x[0], lane); gPH<1>(re, im, 2.0f * x[1], lane);
    gPH<2>(re, im, 2.0f * x[2], lane); gPH<3>(re, im, 2.0f * x[3], lane);
    gPH<4>(re, im, 2.0f * x[4], lane); gPH<5>(re, im, 2.0f * x[5], lane);
    gPH<6>(re, im, 2.0f * x[6], lane); gPH<7>(re, im, 2.0f * x[7], lane);
  }
  convblk<0, 1>(re, im, c1 + 0, lane);
  convblk<2, 3>(re, im, c1 + 3, lane);
  convblk<4, 5>(re, im, c1 + 6, lane);
  convblk<6, 7>(re, im, c1 + 9, lane);
  poolblk<0, 4>(re, im, p1 + 0, lane);
  poolblk<1, 5>(re, im, p1 + 3, lane);
  poolblk<2, 6>(re, im, p1 + 6, lane);
  poolblk<3, 7>(re, im, p1 + 9, lane);
  convblk<0, 1>(re, im, c2 + 0, lane);
  convblk<2, 3>(re, im, c2 + 3, lane);
  poolblk<0, 2>(re, im, p2 + 0, lane);
  poolblk<1, 3>(re, im, p2 + 3, lane);
  convblk<0, 1>(re, im, c3 + 0, lane);
  poolblk<0, 1>(re, im, p3 + 0, lane);

  // <Z> on qubit 7  (bit 0 of r)
  float acc = 0.0f;
  #pragma unroll
  for (int r = 0; r < 8; ++r) {
    float p = re[r] * re[r] + im[r] * im[r];
    acc += (r & 1) ? -p : p;
  }
  #pragma unroll
  for (int off = 16; off >= 1; off >>= 1) acc += shflx(acc, off);
  if (lane == 0) qout[wid] = acc;
}

// ---------------------------------------------------------------------------
// Sampler: 2-qubit real circuit, one thread per batch element
// ---------------------------------------------------------------------------
DEVINL void sry0(float& s00, float& s01, float& s10, float& s11, float t) {
  float s, c; sincosf(0.5f * t, &s, &c);
  float n00 = c * s00 - s * s10, n01 = c * s01 - s * s11;
  float n10 = s * s00 + c * s10, n11 = s * s01 + c * s11;
  s00 = n00; s01 = n01; s10 = n10; s11 = n11;
}
DEVINL void sry1(float& s00, float& s01, float& s10, float& s11, float t) {
  float s, c; sincosf(0.5f * t, &s, &c);
  float n00 = c * s00 - s * s01, n01 = s * s00 + c * s01;
  float n10 = c * s10 - s * s11, n11 = s * s10 + c * s11;
  s00 = n00; s01 = n01; s10 = n10; s11 = n11;
}

__global__ void k_samp(const float* __restrict__ inp, const float* __restrict__ w,
                       float* __restrict__ samp) {
  const int b = blockIdx.x * blockDim.x + threadIdx.x;
  if (b >= NBATCH) return;
  float w0 = w[0], w1 = w[1], w2 = w[2], w3 = w[3];
  float s00 = 1.0f, s01 = 0.0f, s10 = 0.0f, s11 = 0.0f;
  sry0(s00, s01, s10, s11, inp[b * 8 + 0]);
  sry1(s00, s01, s10, s11, inp[b * 8 + 1]);
  { float t = s10; s10 = s11; s11 = t; }        // cnot(0,1)
  sry0(s00, s01, s10, s11, w0);
  sry1(s00, s01, s10, s11, w1);
  { float t = s10; s10 = s11; s11 = t; }
  sry0(s00, s01, s10, s11, w2);
  sry1(s00, s01, s10, s11, w3);
  // probs.T.reshape(4) = [p00, p10, p01, p11]
  samp[b * 4 + 0] = s00 * s00;
  samp[b * 4 + 1] = s10 * s10;
  samp[b * 4 + 2] = s01 * s01;
  samp[b * 4 + 3] = s11 * s11;
}

// ---------------------------------------------------------------------------
// xproj GEMM: (4096x8) @ Wg_x.T (8x16) + bias, 4 gates -> (4096x64), f32 WMMA
// One wave per 16-row tile; 2 K-tiles x 4 gates = 8 v_wmma_f32_16x16x4_f32
// ---------------------------------------------------------------------------
__global__ void k_xproj(const float* __restrict__ inp,
                        const float* __restrict__ Wf, const float* __restrict__ bfv,
                        const float* __restrict__ Wi, const float* __restrict__ biv,
                        const float* __restrict__ Wu, const float* __restrict__ buv,
                        const float* __restrict__ Wo, const float* __restrict__ bov,
                        float* __restrict__ xproj) {
  const int tile = (blockIdx.x * blockDim.x + threadIdx.x) >> 5;
  const int lane = threadIdx.x & 31;
  const int n  = lane & 15;
  const int kh = lane >> 4;
  const int row = tile * 16 + n;

  v2f a0, a1;               // A: 16x4 f32 tiles, K-tile 0 = cols 0..3, tile 1 = cols 4..7
  a0.x = inp[row * 8 + kh * 2 + 0];
  a0.y = inp[row * 8 + kh * 2 + 1];
  a1.x = inp[row * 8 + 4 + kh * 2 + 0];
  a1.y = inp[row * 8 + 4 + kh * 2 + 1];

  const float* Ws[4] = { Wf, Wi, Wu, Wo };
  const float* Bs[4] = { bfv, biv, buv, bov };

  #pragma unroll
  for (int g = 0; g < 4; ++g) {
    const float* W = Ws[g];
    v2f b0, b1;             // B[k][n] = W[n*24 + k]
    b0.x = W[n * 24 + kh * 2 + 0];
    b0.y = W[n * 24 + kh * 2 + 1];
    b1.x = W[n * 24 + 4 + kh * 2 + 0];
    b1.y = W[n * 24 + 4 + kh * 2 + 1];
    v8f c = {};
    c = __builtin_amdgcn_wmma_f32_16x16x4_f32(false, a0, false, b0, (short)0, c, false, false);
    c = __builtin_amdgcn_wmma_f32_16x16x4_f32(false, a1, false, b1, (short)0, c, false, false);
    float bias = Bs[g][n];
    #pragma unroll
    for (int j = 0; j < 8; ++j) {
      const int m = tile * 16 + j + 8 * kh;
      xproj[m * 64 + g * 16 + n] = c[j] + bias;
    }
  }
}

// ---------------------------------------------------------------------------
// Serial QLSTM over 4096 steps: one wave, lane n<16 owns hidden unit n.
// ---------------------------------------------------------------------------
DEVINL float qlayer_lane(float cval, int n) {
  // inclusive cumprod across lanes 0..15
  float cum = cval;
  #pragma unroll
  for (int off = 1; off < 16; off <<= 1) {
    float o = __shfl_up(cum, off, 16);
    cum = (n >= off) ? cum * o : cum;
  }
  // product of c[1..15]
  float p = (n == 0) ? 1.0f : cval;
  #pragma unroll
  for (int m = 1; m < 16; m <<= 1) p *= __shfl_xor(p, m, 16);
  return (n == 0) ? p : cum;
}

DEVINL float sigmoidf(float x) { return 1.0f / (1.0f + expf(-x)); }

__global__ void k_lstm(const float* __restrict__ xproj,
                       const float* __restrict__ Wf, const float* __restrict__ Wi,
                       const float* __restrict__ Wu, const float* __restrict__ Wo,
                       const float* __restrict__ thf, const float* __restrict__ thi,
                       const float* __restrict__ thu, const float* __restrict__ tho,
                       float* __restrict__ houts) {
  const int lane = threadIdx.x;
  const int n = lane & 15;

  float whf[16], whi[16], whu[16], who[16];
  #pragma unroll
  for (int k = 0; k < 16; ++k) {
    whf[k] = Wf[n * 24 + 8 + k];
    whi[k] = Wi[n * 24 + 8 + k];
    whu[k] = Wu[n * 24 + 8 + k];
    who[k] = Wo[n * 24 + 8 + k];
  }
  const float tf = thf[n], ti = thi[n], tu = thu[n], to = tho[n];

  float h = 0.0f, cst = 0.0f;
  for (int t = 0; t < NBATCH; ++t) {
    const int tp = (t + 8 < NBATCH) ? (t + 8) : (NBATCH - 1);
    __builtin_prefetch(&xproj[tp * 64 + lane], 0, 1);   // global_prefetch_b8

    float pf = xproj[t * 64 +  0 + n];
    float pi = xproj[t * 64 + 16 + n];
    float pu = xproj[t * 64 + 32 + n];
    float po = xproj[t * 64 + 48 + n];
    #pragma unroll
    for (int k = 0; k < 16; ++k) {
      float hk = __shfl(h, k, 32);
      pf = fmaf(hk, whf[k], pf);
      pi = fmaf(hk, whi[k], pi);
      pu = fmaf(hk, whu[k], pu);
      po = fmaf(hk, who[k], po);
    }
    float f = sigmoidf(qlayer_lane(cosf(pf + tf), n));
    float i = sigmoidf(qlayer_lane(cosf(pi + ti), n));
    float g = tanhf(qlayer_lane(cosf(pu + tu), n));
    float o = sigmoidf(qlayer_lane(cosf(po + to), n));
    cst = f * cst + i * g;
    h = o * tanhf(cst);
    if (lane < 16) houts[t * 16 + n] = h;
  }
}

// ---------------------------------------------------------------------------
// Tag head: (4096x16) @ Wt.T (16x10, padded to 16) + bt, then log_softmax(10)
// One wave per 16-row tile, 4 K-tiles = 4 v_wmma_f32_16x16x4_f32
// ---------------------------------------------------------------------------
__global__ void k_tag(const float* __restrict__ houts,
                      const float* __restrict__ Wt, const float* __restrict__ bt,
                      float* __restrict__ tag) {
  const int tile = (blockIdx.x * blockDim.x + threadIdx.x) >> 5;
  const int lane = threadIdx.x & 31;
  const int n  = lane & 15;
  const int kh = lane >> 4;
  const int row = tile * 16 + n;
  const bool valid = (n < 10);

  v8f c = {};
  #pragma unroll
  for (int kt = 0; kt < 4; ++kt) {
    v2f av, bv;
    av.x = houts[row * 16 + kt * 4 + kh * 2 + 0];
    av.y = houts[row * 16 + kt * 4 + kh * 2 + 1];
    bv.x = valid ? Wt[n * 16 + kt * 4 + kh * 2 + 0] : 0.0f;
    bv.y = valid ? Wt[n * 16 + kt * 4 + kh * 2 + 1] : 0.0f;
    c = __builtin_amdgcn_wmma_f32_16x16x4_f32(false, av, false, bv, (short)0, c, false, false);
  }
  const float bias = valid ? bt[n] : 0.0f;

  #pragma unroll
  for (int j = 0; j < 8; ++j) {
    float logit = c[j] + bias;
    float v = valid ? logit : -3.402823466e38f;
    #pragma unroll
    for (int m = 1; m < 16; m <<= 1) v = fmaxf(v, __shfl_xor(v, m, 16));
    float e = valid ? expf(logit - v) : 0.0f;
    #pragma unroll
    for (int m = 1; m < 16; m <<= 1) e += __shfl_xor(e, m, 16);
    if (valid) {
      const int r = tile * 16 + j + 8 * kh;
      tag[r * 10 + n] = logit - v - logf(e);
    }
  }
}

// ---------------------------------------------------------------------------
extern "C" void kernel_launch(void* const* d_in, const int* in_sizes, int n_in,
                              void* d_out, int out_size, void* d_ws, size_t ws_size,
                              hipStream_t stream) {
  const float* inputs = (const float*)d_in[0];
  const float* c1 = (const float*)d_in[1];
  const float* p1 = (const float*)d_in[2];
  const float* c2 = (const float*)d_in[3];
  const float* p2 = (const float*)d_in[4];
  const float* c3 = (const float*)d_in[5];
  const float* p3 = (const float*)d_in[6];
  const float* w_smp = (const float*)d_in[7];
  const float* Wf = (const float*)d_in[8];
  const float* bf = (const float*)d_in[9];
  const float* Wi = (const float*)d_in[10];
  const float* bi = (const float*)d_in[11];
  const float* Wu = (const float*)d_in[12];
  const float* bu = (const float*)d_in[13];
  const float* Wo = (const float*)d_in[14];
  const float* bo = (const float*)d_in[15];
  const float* th_f = (const float*)d_in[16];
  const float* th_i = (const float*)d_in[17];
  const float* th_u = (const float*)d_in[18];
  const float* th_o = (const float*)d_in[19];
  const float* Wt = (const float*)d_in[20];
  const float* bt = (const float*)d_in[21];

  float* out = (float*)d_out;
  float* qcnn_out = out;                         // 4096
  float* tag_out  = out + NBATCH;                // 40960
  float* samp_out = out + NBATCH + NBATCH * 10;  // 16384

  float* xproj = (float*)d_ws;                   // 4096*64 f32 = 1 MB
  float* houts = xproj + NBATCH * 64;            // 4096*16 f32 = 256 KB

  // Batch-parallel parts
  k_qcnn<<<512, 256, 0, stream>>>(inputs, c1, p1, c2, p2, c3, p3, qcnn_out);
  k_samp<<<16, 256, 0, stream>>>(inputs, w_smp, samp_out);

  // LSTM: parallel x-projection (WMMA) -> serial recurrence -> parallel head (WMMA)
  k_xproj<<<32, 256, 0, stream>>>(inputs, Wf, bf, Wi, bi, Wu, bu, Wo, bo, xproj);
  k_lstm<<<1, 32, 0, stream>>>(xproj, Wf, Wi, Wu, Wo, th_f, th_i, th_u, th_o, houts);
  k_tag<<<32, 256, 0, stream>>>(houts, Wt, bt, tag_out);
}